// PillarFeatureNet_58171037057706
// MI455X (gfx1250) — compile-verified
//
#include <hip/hip_runtime.h>

typedef __attribute__((ext_vector_type(2))) float v2f;
typedef __attribute__((ext_vector_type(8))) float v8f;

#define RESF  0.16f
#define XMINF -51.2f
#define YMINF -51.2f
#define EPSF  1e-5f
#define NEGV  -1000000000.0f
#define WPB   8   // waves per block (256 threads)

// Cross-half swap (lane i <-> lane i^16) as a pure-VALU v_permlanex16_b32
// with identity lane selects. Avoids ds_bpermute + s_wait_dscnt on the
// critical path between WMMA issues (VALU co-executes with the XDL pipe).
__device__ __forceinline__ float xswap16(float x) {
    int v = __float_as_int(x);
    v = __builtin_amdgcn_permlanex16(v, v, 0x76543210, (int)0xfedcba98, false, false);
    return __int_as_float(v);
}

__global__ __launch_bounds__(256, 1)
void pfn_kernel(const float* __restrict__ pillars,
                const long long* __restrict__ coords,
                const long long* __restrict__ npts,
                const float* __restrict__ W1, const float* __restrict__ b1,
                const float* __restrict__ g1, const float* __restrict__ beta1,
                const float* __restrict__ m1, const float* __restrict__ v1,
                const float* __restrict__ W2, const float* __restrict__ b2v,
                const float* __restrict__ g2, const float* __restrict__ beta2,
                const float* __restrict__ m2, const float* __restrict__ v2,
                float* __restrict__ out, int P)
{
    __shared__ float sW1s[12 * 64];        // s1-scaled W1, K padded 9->12 with zeros
    __shared__ float sU1[64];              // fused bias/BN1 shift
    __shared__ float sFeat[WPB][32 * 13];  // per-wave features, stride 13 (bank-friendly)

    const int tid = threadIdx.x;
    // Block-cooperative constant tables (built once, reused for all pillars)
    for (int i = tid; i < 64; i += 256) {
        float s1 = g1[i] * rsqrtf(v1[i] + EPSF);
        sU1[i] = s1 * b1[i] + beta1[i] - s1 * m1[i];
    }
    for (int i = tid; i < 12 * 64; i += 256) {
        int kk = i >> 6, c = i & 63;
        float s1 = g1[c] * rsqrtf(v1[c] + EPSF);
        sW1s[i] = (kk < 9) ? W1[kk * 64 + c] * s1 : 0.0f;
    }
    __syncthreads();

    const int lane = tid & 31;
    const int wave = tid >> 5;
    const int l15  = lane & 15;
    const int hi   = lane >> 4;   // which half of the wave
    const int h2   = hi * 2;
    const bool low = (hi == 0);

    // BN2 fused constants, chan = 16n + l15 (same layout as D2 columns)
    float s2v[4], t2v[4];
#pragma unroll
    for (int n = 0; n < 4; ++n) {
        int c = 16 * n + l15;
        float s = g2[c] * rsqrtf(v2[c] + EPSF);
        s2v[n] = s;
        t2v[n] = s * b2v[c] + beta2[c] - s * m2[c];
    }

    // Resident B-fragments of W2 (column-scaled by s2): B-layout == A-layout of W2^T.
    // 4 n-tiles x 16 K-steps x 2 floats = 128 VGPRs, loaded once per wave and
    // amortized over the grid-stride pillar loop.
    v2f b2f[4][16];
#pragma unroll
    for (int k = 0; k < 16; ++k)
#pragma unroll
        for (int n = 0; n < 4; ++n) {
            int kk0 = 4 * k + h2;
            int c   = 16 * n + l15;
            v2f t;
            t[0] = W2[kk0 * 64 + c]       * s2v[n];
            t[1] = W2[(kk0 + 1) * 64 + c] * s2v[n];
            b2f[n][k] = t;
        }

    const int nw = gridDim.x * WPB;
    float* feat = &sFeat[wave][0];

    for (int p = blockIdx.x * WPB + wave; p < P; p += nw) {
        // Raw point data: point == lane, 512B coalesced per wave
        const float4 pt = reinterpret_cast<const float4*>(pillars)[p * 32 + lane];
        if (p + nw < P)
            __builtin_prefetch(&pillars[(size_t)(p + nw) * 128 + lane * 4], 0, 1);
        const int np = (int)npts[p];
        const float xc = ((float)(int)coords[2 * p + 1] + 0.5f) * RESF + XMINF;
        const float yc = ((float)(int)coords[2 * p + 0] + 0.5f) * RESF + YMINF;

        // masked z-mean via wave reduction
        float zm = (lane < np) ? pt.z : 0.0f;
        zm += xswap16(zm);
#pragma unroll
        for (int s = 8; s > 0; s >>= 1) zm += __shfl_xor(zm, s, 32);
        zm /= fmaxf((float)np, 1.0f);

        const float xo = pt.x - xc, yo = pt.y - yc, zo = pt.z - zm;
        float* fr = feat + lane * 13;
        fr[0] = pt.x; fr[1] = pt.y; fr[2] = pt.z; fr[3] = pt.w;
        fr[4] = xo;   fr[5] = yo;   fr[6] = zo;   fr[7] = xo;  fr[8] = yo;
        fr[9] = 0.0f; fr[10] = 0.0f; fr[11] = 0.0f;
        __builtin_amdgcn_wave_barrier();

        v8f d2[2][4];
#pragma unroll
        for (int m = 0; m < 2; ++m)
#pragma unroll
            for (int n = 0; n < 4; ++n)
#pragma unroll
                for (int r = 0; r < 8; ++r) d2[m][n][r] = 0.0f;

        // Interleaved: produce one 16-chan block of h1^T, consume it in 4 K-steps of GEMM2
#pragma unroll
        for (int mt = 0; mt < 4; ++mt) {
            // A1 fragments: (s1*W1)^T rows 16mt..16mt+15 (transient, from LDS)
            v2f a1f[3];
#pragma unroll
            for (int k = 0; k < 3; ++k) {
                int base = (4 * k + h2) * 64 + 16 * mt + l15;
                v2f t; t[0] = sW1s[base]; t[1] = sW1s[base + 64];
                a1f[k] = t;
            }
            // B1 fragments: feat^T column-major == feat row-major in A-layout
            // (transient per mt to keep peak VGPR pressure down)
            v2f b1f[2][3];
#pragma unroll
            for (int nt = 0; nt < 2; ++nt)
#pragma unroll
                for (int k = 0; k < 3; ++k) {
                    const float* q = feat + (16 * nt + l15) * 13 + 4 * k + h2;
                    v2f t; t[0] = q[0]; t[1] = q[1];
                    b1f[nt][k] = t;
                }
            // fused bias as WMMA C operand: element (r, lane) -> chan 16mt+r+8*hi
            v8f c1;
#pragma unroll
            for (int r = 0; r < 8; ++r) c1[r] = sU1[16 * mt + r + 8 * hi];

            v8f d1[2];
#pragma unroll
            for (int nt = 0; nt < 2; ++nt) {
                v8f acc = c1;
#pragma unroll
                for (int k = 0; k < 3; ++k)
                    acc = __builtin_amdgcn_wmma_f32_16x16x4_f32(
                        false, a1f[k], false, b1f[nt][k], (short)0, acc, false, false);
#pragma unroll
                for (int r = 0; r < 8; ++r) acc[r] = fmaxf(acc[r], 0.0f);  // BN1+ReLU
                d1[nt] = acc;
            }

            // GEMM2 partial: global K-steps 4*mt..4*mt+3.
            // A2 fragment (points on rows) built from D1 (points on lanes):
            // exactly one VALU half-swap + one select per register.
#pragma unroll
            for (int kk = 0; kk < 4; ++kk) {
                v2f a2f[2];
#pragma unroll
                for (int m = 0; m < 2; ++m) {
                    v2f t;
#pragma unroll
                    for (int j = 0; j < 2; ++j) {
                        int cl_lo = 4 * kk + j;   // chan_local wanted by lower-half lanes
                        int cl_hi = cl_lo + 2;    // chan_local wanted by upper-half lanes
                        float vlo, vhi;
                        if (cl_lo < 8) {          // kk=0,1: lo direct, hi cross-half
                            vlo = d1[m][cl_lo];
                            vhi = xswap16(d1[m][cl_hi]);
                        } else {                  // kk=2,3: lo cross-half, hi direct
                            vlo = xswap16(d1[m][cl_lo - 8]);
                            vhi = d1[m][cl_hi - 8];
                        }
                        t[j] = low ? vlo : vhi;
                    }
                    a2f[m] = t;
                }
#pragma unroll
                for (int m = 0; m < 2; ++m)
#pragma unroll
                    for (int n = 0; n < 4; ++n)
                        d2[m][n] = __builtin_amdgcn_wmma_f32_16x16x4_f32(
                            false, a2f[m], false, b2f[n][4 * mt + kk], (short)0,
                            d2[m][n], false, false);
            }
        }

        // Epilogue: +t2 (s2 pre-folded into b2f), ReLU, validity mask, max over 32 points
        float red[4];
#pragma unroll
        for (int n = 0; n < 4; ++n) {
            float mx = NEGV;
#pragma unroll
            for (int m = 0; m < 2; ++m)
#pragma unroll
                for (int r = 0; r < 8; ++r) {
                    float val = fmaxf(d2[m][n][r] + t2v[n], 0.0f);
                    int pidx = 16 * m + r + 8 * hi;
                    val = (pidx < np) ? val : NEGV;
                    mx = fmaxf(mx, val);
                }
            mx = fmaxf(mx, xswap16(mx));  // combine point halves
            red[n] = mx;
        }
        // chan == lane for o0 (tiles 0/1), chan == 32+lane for o1 (tiles 2/3)
        float o0 = low ? red[0] : red[1];
        float o1 = low ? red[2] : red[3];
        out[(size_t)p * 64 + lane]      = o0;
        out[(size_t)p * 64 + 32 + lane] = o1;
    }
}

extern "C" void kernel_launch(void* const* d_in, const int* in_sizes, int n_in,
                              void* d_out, int out_size, void* d_ws, size_t ws_size,
                              hipStream_t stream) {
    const float*     pillars = (const float*)d_in[0];
    const long long* coords  = (const long long*)d_in[1];
    const long long* npts    = (const long long*)d_in[2];
    const float* W1    = (const float*)d_in[3];
    const float* b1    = (const float*)d_in[4];
    const float* g1    = (const float*)d_in[5];
    const float* beta1 = (const float*)d_in[6];
    const float* m1    = (const float*)d_in[7];
    const float* v1    = (const float*)d_in[8];
    const float* W2    = (const float*)d_in[9];
    const float* b2    = (const float*)d_in[10];
    const float* g2    = (const float*)d_in[11];
    const float* beta2 = (const float*)d_in[12];
    const float* m2    = (const float*)d_in[13];
    const float* v2    = (const float*)d_in[14];

    int P = in_sizes[0] / (32 * 4);
    int blocks = (P + WPB - 1) / WPB;
    if (blocks > 1024) blocks = 1024;   // grid-stride: amortize resident W2 fragments

    pfn_kernel<<<blocks, 256, 0, stream>>>(pillars, coords, npts,
                                           W1, b1, g1, beta1, m1, v1,
                                           W2, b2, g2, beta2, m2, v2,
                                           (float*)d_out, P);
}